// GATDensityRegressor_78649441124582
// MI455X (gfx1250) — compile-verified
//
#include <hip/hip_runtime.h>
#include <math.h>

#define N_NODES 50000
#define N_EDGES 800000
#define E_TOT   (N_EDGES + N_NODES)   // 850000, self-loops appended
#define HEADS   4
#define HID     48
#define FDIM    (HEADS * HID)         // 192
#define NEG_SLOPE 0.2f

typedef __attribute__((ext_vector_type(2))) float v2f;
typedef __attribute__((ext_vector_type(8))) float v8f;

// ---- monotone float <-> uint encoding so atomicMax(uint) == float max ----
__device__ __forceinline__ unsigned fenc(float f) {
  unsigned u = __float_as_uint(f);
  return (u & 0x80000000u) ? ~u : (u | 0x80000000u);
}
__device__ __forceinline__ float fdec(unsigned u) {
  return (u & 0x80000000u) ? __uint_as_float(u & 0x7FFFFFFFu) : __uint_as_float(~u);
}

// =====================================================================
// Layer 0 linear: x[N,3] @ W[3,192] + b  (both Wl and Wr in one pass)
// =====================================================================
__global__ void linear_din3_kernel(const float* __restrict__ x,
                                   const float* __restrict__ Wl, const float* __restrict__ bl,
                                   const float* __restrict__ Wr, const float* __restrict__ br,
                                   float* __restrict__ xl, float* __restrict__ xr) {
  int idx = blockIdx.x * blockDim.x + threadIdx.x;
  if (idx >= N_NODES * FDIM) return;
  int n = idx / FDIM;
  int c = idx - n * FDIM;
  float x0 = x[n * 3 + 0], x1 = x[n * 3 + 1], x2 = x[n * 3 + 2];
  xl[idx] = bl[c] + x0 * Wl[c] + x1 * Wl[FDIM + c] + x2 * Wl[2 * FDIM + c];
  xr[idx] = br[c] + x0 * Wr[c] + x1 * Wr[FDIM + c] + x2 * Wr[2 * FDIM + c];
}

// =====================================================================
// Layers 1/2 linear: X[N,48] @ W[48,192] + b via V_WMMA_F32_16X16X4_F32
// One wave (32 lanes) computes one 16x16 tile; K=48 -> 12 WMMA ops.
// A 16x4 layout: lanes 0-15 hold {K=0,K=1}, lanes 16-31 hold {K=2,K=3}.
// B 4x16 layout: lanes 0-15 hold rows {K=0,K=1} at N=lane, lanes 16-31
//                hold rows {K=2,K=3} at N=lane-16.
// C/D:  vgpr i -> M=i (lanes 0-15) / M=i+8 (lanes 16-31), N = lane&15.
// =====================================================================
__global__ __launch_bounds__(32) void gemm48x192_wmma_kernel(
    const float* __restrict__ X, const float* __restrict__ W,
    const float* __restrict__ bias, float* __restrict__ Y) {
  const int lane = threadIdx.x;
  const int half = lane >> 4;       // 0 or 1
  const int l16  = lane & 15;
  const int m0   = blockIdx.x * 16;
  const int n0   = blockIdx.y * 16;
  const int arow = m0 + l16;
  const int bcol = n0 + l16;
  v8f acc = {};
#pragma unroll
  for (int kk = 0; kk < HID; kk += 4) {
    const int ka = kk + half * 2;
    v2f a, b;
    a.x = X[arow * HID + ka];
    a.y = X[arow * HID + ka + 1];
    b.x = W[ka * FDIM + bcol];
    b.y = W[(ka + 1) * FDIM + bcol];
    acc = __builtin_amdgcn_wmma_f32_16x16x4_f32(false, a, false, b,
                                                (short)0, acc, false, false);
  }
  const float bn = bias[bcol];
#pragma unroll
  for (int i = 0; i < 8; ++i) {
    const int m = m0 + i + half * 8;
    Y[m * FDIM + bcol] = acc[i] + bn;
  }
}

// =====================================================================
// Edge scoring: one wave per edge. 192 channels = 32 lanes x 6.
// score[e,h] = sum_c att[h,c] * leaky_relu(xl[src,h,c] + xr[dst,h,c])
// Also segment-max via encoded atomicMax on maxenc[dst,h].
// =====================================================================
__global__ void edge_score_kernel(const long long* __restrict__ ei,
                                  const float* __restrict__ xl,
                                  const float* __restrict__ xr,
                                  const float* __restrict__ att,
                                  float* __restrict__ score,
                                  unsigned* __restrict__ maxenc) {
  const int e = (blockIdx.x * blockDim.x + threadIdx.x) >> 5;
  const int lane = threadIdx.x & 31;
  if (e >= E_TOT) return;
  int src, dst;
  if (e < N_EDGES) { src = (int)ei[e]; dst = (int)ei[N_EDGES + e]; }
  else             { src = e - N_EDGES; dst = src; }
  const int base = lane * 6;                       // stays inside one head (48 = 8*6)
  const float* pl = xl + (size_t)src * FDIM + base;
  const float* pr = xr + (size_t)dst * FDIM + base;
  float partial = 0.f;
#pragma unroll
  for (int j = 0; j < 6; ++j) {
    float v = pl[j] + pr[j];
    v = (v > 0.f) ? v : v * NEG_SLOPE;
    partial += att[base + j] * v;
  }
  // reduce over the 8 lanes covering one head
  partial += __shfl_xor(partial, 1, 32);
  partial += __shfl_xor(partial, 2, 32);
  partial += __shfl_xor(partial, 4, 32);
  if ((lane & 7) == 0) {
    const int h = lane >> 3;
    score[(size_t)e * HEADS + h] = partial;
    atomicMax(&maxenc[dst * HEADS + h], fenc(partial));
  }
}

// p = exp(score - max); denom[dst,h] += p
__global__ void edge_exp_kernel(const long long* __restrict__ ei,
                                float* __restrict__ score,
                                const unsigned* __restrict__ maxenc,
                                float* __restrict__ denom) {
  const int idx = blockIdx.x * blockDim.x + threadIdx.x;
  if (idx >= E_TOT * HEADS) return;
  const int e = idx >> 2, h = idx & 3;
  const int dst = (e < N_EDGES) ? (int)ei[N_EDGES + e] : (e - N_EDGES);
  const float m = fdec(maxenc[dst * HEADS + h]);
  const float p = expf(score[idx] - m);
  score[idx] = p;
  atomicAdd(&denom[dst * HEADS + h], p);
}

// alpha = p / (denom + 1e-16), folding the 1/HEADS head-mean
__global__ void edge_alpha_kernel(const long long* __restrict__ ei,
                                  float* __restrict__ score,
                                  const float* __restrict__ denom) {
  const int idx = blockIdx.x * blockDim.x + threadIdx.x;
  if (idx >= E_TOT * HEADS) return;
  const int e = idx >> 2, h = idx & 3;
  const int dst = (e < N_EDGES) ? (int)ei[N_EDGES + e] : (e - N_EDGES);
  score[idx] = score[idx] / (denom[dst * HEADS + h] + 1e-16f) * (1.0f / HEADS);
}

// hout[dst,c] += sum_h alpha[e,h] * xl[src,h,c]   (mean folded into alpha)
__global__ void edge_agg_kernel(const long long* __restrict__ ei,
                                const float* __restrict__ alpha,
                                const float* __restrict__ xl,
                                float* __restrict__ hout) {
  const int idx = blockIdx.x * blockDim.x + threadIdx.x;
  if (idx >= E_TOT * HID) return;
  const int e = idx / HID;
  const int c = idx - e * HID;
  int src, dst;
  if (e < N_EDGES) { src = (int)ei[e]; dst = (int)ei[N_EDGES + e]; }
  else             { src = e - N_EDGES; dst = src; }
  const float* px = xl + (size_t)src * FDIM + c;
  const float* pa = alpha + (size_t)e * HEADS;
  const float v = pa[0] * px[0] + pa[1] * px[HID] + pa[2] * px[2 * HID] + pa[3] * px[3 * HID];
  atomicAdd(&hout[(size_t)dst * HID + c], v);
}

// x = elu(hout + bias)
__global__ void bias_elu_kernel(const float* __restrict__ hout,
                                const float* __restrict__ bias,
                                float* __restrict__ hin) {
  const int idx = blockIdx.x * blockDim.x + threadIdx.x;
  if (idx >= N_NODES * HID) return;
  const int c = idx % HID;
  const float v = hout[idx] + bias[c];
  hin[idx] = (v > 0.f) ? v : expm1f(v);
}

// out = relu(x @ head_w + head_b)
__global__ void head_kernel(const float* __restrict__ hin,
                            const float* __restrict__ hw,
                            const float* __restrict__ hb,
                            float* __restrict__ out) {
  const int n = blockIdx.x * blockDim.x + threadIdx.x;
  if (n >= N_NODES) return;
  float s = hb[0];
#pragma unroll
  for (int c = 0; c < HID; ++c) s += hin[n * HID + c] * hw[c];
  out[n] = (s > 0.f) ? s : 0.f;
}

// =====================================================================
extern "C" void kernel_launch(void* const* d_in, const int* in_sizes, int n_in,
                              void* d_out, int out_size, void* d_ws, size_t ws_size,
                              hipStream_t stream) {
  const float* x = (const float*)d_in[0];
  const long long* ei = (const long long*)d_in[1];   // int64 (2, N_EDGES)
  const float *Wl[3], *bl[3], *Wr[3], *br[3], *att[3], *bias[3];
  for (int li = 0; li < 3; ++li) {
    Wl[li]   = (const float*)d_in[2 + li * 6 + 0];
    bl[li]   = (const float*)d_in[2 + li * 6 + 1];
    Wr[li]   = (const float*)d_in[2 + li * 6 + 2];
    br[li]   = (const float*)d_in[2 + li * 6 + 3];
    att[li]  = (const float*)d_in[2 + li * 6 + 4];
    bias[li] = (const float*)d_in[2 + li * 6 + 5];
  }
  const float* head_w = (const float*)d_in[20];
  const float* head_b = (const float*)d_in[21];
  float* out = (float*)d_out;

  // carve workspace (~111 MB)
  char* p = (char*)d_ws;
  auto carve = [&](size_t bytes) -> char* {
    char* r = p;
    p += (bytes + 255) & ~(size_t)255;
    return r;
  };
  float*    xl   = (float*)   carve((size_t)N_NODES * FDIM * sizeof(float));
  float*    xr   = (float*)   carve((size_t)N_NODES * FDIM * sizeof(float));
  float*    sc   = (float*)   carve((size_t)E_TOT * HEADS * sizeof(float));
  unsigned* mx   = (unsigned*)carve((size_t)N_NODES * HEADS * sizeof(unsigned));
  float*    dn   = (float*)   carve((size_t)N_NODES * HEADS * sizeof(float));
  float*    hin  = (float*)   carve((size_t)N_NODES * HID * sizeof(float));
  float*    hout = (float*)   carve((size_t)N_NODES * HID * sizeof(float));

  const int B = 256;
  for (int li = 0; li < 3; ++li) {
    if (li == 0) {
      const int tot = N_NODES * FDIM;
      linear_din3_kernel<<<(tot + B - 1) / B, B, 0, stream>>>(
          x, Wl[0], bl[0], Wr[0], br[0], xl, xr);
    } else {
      dim3 g(N_NODES / 16, FDIM / 16);   // 3125 x 12, exact
      gemm48x192_wmma_kernel<<<g, 32, 0, stream>>>(hin, Wl[li], bl[li], xl);
      gemm48x192_wmma_kernel<<<g, 32, 0, stream>>>(hin, Wr[li], br[li], xr);
    }
    hipMemsetAsync(mx, 0, (size_t)N_NODES * HEADS * sizeof(unsigned), stream); // == -inf encoded
    hipMemsetAsync(dn, 0, (size_t)N_NODES * HEADS * sizeof(float), stream);
    edge_score_kernel<<<(E_TOT + 7) / 8, B, 0, stream>>>(ei, xl, xr, att[li], sc, mx);
    edge_exp_kernel<<<(E_TOT * HEADS + B - 1) / B, B, 0, stream>>>(ei, sc, mx, dn);
    edge_alpha_kernel<<<(E_TOT * HEADS + B - 1) / B, B, 0, stream>>>(ei, sc, dn);
    hipMemsetAsync(hout, 0, (size_t)N_NODES * HID * sizeof(float), stream);
    edge_agg_kernel<<<(E_TOT * HID + B - 1) / B, B, 0, stream>>>(ei, sc, xl, hout);
    bias_elu_kernel<<<(N_NODES * HID + B - 1) / B, B, 0, stream>>>(hout, bias[li], hin);
  }
  head_kernel<<<(N_NODES + B - 1) / B, B, 0, stream>>>(hin, head_w, head_b, out);
}